// SplitSpatialBlock_24816321036292
// MI455X (gfx1250) — compile-verified
//
#include <hip/hip_runtime.h>
#include <math.h>

// ---------------------------------------------------------------------------
// SplitSpatialBlock forward for MI455X (gfx1250, wave32).
// All GEMMs use V_WMMA_F32_16X16X4_F32 (full-f32 matrix core path).
// ---------------------------------------------------------------------------

typedef float v2f __attribute__((ext_vector_type(2)));
typedef float v8f __attribute__((ext_vector_type(8)));

#define DIMC   128   // channel dim
#define DINC   256   // expanded inner dim
#define NSTATE 16
#define NJ     17
#define NBF    972   // B*F = 4*243
#define NF     243
#define NBATCH 4
#define NTOK_T 16524 // 68*243
#define NTPAD  16528 // padded to multiple of 16

__device__ __forceinline__ float siluf(float x) { return x / (1.f + expf(-x)); }
__device__ __forceinline__ float geluf(float x) { return 0.5f * x * (1.f + erff(x * 0.70710678118654752f)); }
__device__ __forceinline__ float softplusf(float x) { return (x > 20.f) ? x : log1pf(expf(x)); }

// ---------------------------------------------------------------------------
// WMMA GEMM: C(M,N) = act( A(M,K; lda) @ W(N,K; ldw)^T + bias )
// One wave computes a 16x32 tile (two 16x16 N-tiles sharing one A fragment).
// K must be a multiple of 4.
//
// Out-of-range rows/cols are handled by CLAMPING the source row index
// (A row r only affects D row r; W row n only affects D col n; padded
// rows/cols are never stored, so their fragment data may be garbage as long
// as the address is in-bounds). This keeps the inner loop branch-free:
// unconditional float2 loads (global_load_b64) + 2 independent WMMAs.
//
// A fragment (16x4 f32): lane row = lane&15, k offset = 2*(lane>>4), float2.
// B fragment (4x16 f32): lane col = lane&15, same K split (B = W^T).
// C/D: 8 VGPRs, row = tile_m + v + 8*(lane>>4), col = tile_n + (lane&15).
// ACT: 0=none, 1=softplus, 2=gelu(exact)
// ---------------------------------------------------------------------------
template <int ACT>
__global__ void wmma_gemm_kernel(const float* __restrict__ A, int lda,
                                 const float* __restrict__ W, int ldw,
                                 const float* __restrict__ bias,
                                 float* __restrict__ C, int ldc,
                                 int M, int N, int K, int ntiles2)
{
    int wave = blockIdx.x * (blockDim.x >> 5) + (threadIdx.x >> 5);
    int tm  = wave / ntiles2;
    int tn2 = wave - tm * ntiles2;       // 32-wide column-pair index
    if (tm * 16 >= M) return;            // wave-uniform exit (EXEC stays full)

    int lane = threadIdx.x & 31;
    int hi   = lane >> 4;                // 0: K pair {0,1}, 1: K pair {2,3}
    int ln   = lane & 15;

    int arow = tm * 16 + ln;        if (arow >= M) arow = M - 1;   // clamp
    int col0 = tn2 * 32 + ln;
    int col1 = col0 + 16;
    int wr0  = col0 < N ? col0 : N - 1;                            // clamp
    int wr1  = col1 < N ? col1 : N - 1;                            // clamp

    const float* Ap  = A + (size_t)arow * lda + 2 * hi;
    const float* Wp0 = W + (size_t)wr0 * ldw + 2 * hi;
    const float* Wp1 = W + (size_t)wr1 * ldw + 2 * hi;

    v8f acc0 = {}, acc1 = {};
#pragma unroll 4
    for (int k = 0; k < K; k += 4) {
        v2f a  = *(const v2f*)(Ap  + k);
        v2f b0 = *(const v2f*)(Wp0 + k);
        v2f b1 = *(const v2f*)(Wp1 + k);
        acc0 = __builtin_amdgcn_wmma_f32_16x16x4_f32(
            false, a, false, b0, (short)0, acc0, false, false);
        acc1 = __builtin_amdgcn_wmma_f32_16x16x4_f32(
            false, a, false, b1, (short)0, acc1, false, false);
    }

    int rbase = tm * 16 + 8 * hi;
    if (col0 < N) {
        float bv = bias ? bias[col0] : 0.f;
#pragma unroll
        for (int v = 0; v < 8; ++v) {
            int row = rbase + v;
            if (row < M) {
                float r = acc0[v] + bv;
                if (ACT == 1) r = softplusf(r);
                else if (ACT == 2) r = geluf(r);
                C[(size_t)row * ldc + col0] = r;
            }
        }
    }
    if (col1 < N) {
        float bv = bias ? bias[col1] : 0.f;
#pragma unroll
        for (int v = 0; v < 8; ++v) {
            int row = rbase + v;
            if (row < M) {
                float r = acc1[v] + bv;
                if (ACT == 1) r = softplusf(r);
                else if (ACT == 2) r = geluf(r);
                C[(size_t)row * ldc + col1] = r;
            }
        }
    }
}

// ---------------------------------------------------------------------------
// LayerNorm over 128 channels. mode==0: gather limb tokens from x (Nb,17,128)
// using joint mapping; mode==1: contiguous rows of src. One 128-thread block
// per token, LDS tree reduction.
// ---------------------------------------------------------------------------
__global__ void ln_kernel(const float* __restrict__ src,
                          const float* __restrict__ gam,
                          const float* __restrict__ bet,
                          float* __restrict__ dst,
                          int L, int jbase, int spine, int mode, int Ntok)
{
    int tok = blockIdx.x;
    if (tok >= Ntok) return;
    int c = threadIdx.x;

    int row;
    if (mode == 1) {
        row = tok;
    } else {
        int n = tok / L, l = tok - n * L;
        int joint = spine ? (l == 0 ? 0 : 6 + l) : (jbase + l);
        row = n * NJ + joint;
    }
    float v = src[(size_t)row * DIMC + c];

    __shared__ float red[DIMC];
    red[c] = v; __syncthreads();
    for (int s = DIMC / 2; s > 0; s >>= 1) { if (c < s) red[c] += red[c + s]; __syncthreads(); }
    float mean = red[0] * (1.f / DIMC);
    __syncthreads();
    float d = v - mean;
    red[c] = d * d; __syncthreads();
    for (int s = DIMC / 2; s > 0; s >>= 1) { if (c < s) red[c] += red[c + s]; __syncthreads(); }
    float var = red[0] * (1.f / DIMC);

    dst[(size_t)tok * DIMC + c] = d * rsqrtf(var + 1e-5f) * gam[c] + bet[c];
}

// Causal depthwise conv along L + SiLU. xs = xz[:, :256].
__global__ void conv_silu_kernel(const float* __restrict__ xz,     // (Ntok,512)
                                 const float* __restrict__ conv_w, // (256,dconv)
                                 const float* __restrict__ conv_b, // (256)
                                 float* __restrict__ xc,           // (Ntok,256)
                                 int L, int dconv, int Ntok)
{
    int idx = blockIdx.x * blockDim.x + threadIdx.x;
    if (idx >= Ntok * DINC) return;
    int tok = idx >> 8, d = idx & 255;
    int n = tok / L, l = tok - n * L;
    float acc = conv_b[d];
    for (int k = 0; k < dconv; ++k) {
        int ll = l + k - (dconv - 1);
        if (ll >= 0) acc += conv_w[d * dconv + k] * xz[(size_t)(n * L + ll) * 512 + d];
    }
    xc[idx] = siluf(acc);
}

// Selective scan: one thread per (batch, d). 16 states in registers.
// Output already includes +xc*Dp and *silu(z).
__global__ void scan_kernel(const float* __restrict__ dt,    // (Ntok,256)
                            const float* __restrict__ dbl,   // (Ntok,48), B@8, C@24
                            const float* __restrict__ xc,    // (Ntok,256)
                            const float* __restrict__ xz,    // (Ntok,512), z @256
                            const float* __restrict__ A_log, // (256,16)
                            const float* __restrict__ Dp,    // (256)
                            float* __restrict__ ys,          // (Ntok,256)
                            int L, int NB)
{
    int idx = blockIdx.x * blockDim.x + threadIdx.x;
    if (idx >= NB * DINC) return;
    int batch = idx >> 8, d = idx & 255;

    float a[NSTATE], h[NSTATE];
#pragma unroll
    for (int n = 0; n < NSTATE; ++n) { a[n] = -expf(A_log[d * NSTATE + n]); h[n] = 0.f; }
    float Dd = Dp[d];

    for (int l = 0; l < L; ++l) {
        int tok = batch * L + l;
        float dtv = dt[(size_t)tok * DINC + d];
        float xv  = xc[(size_t)tok * DINC + d];
        const float* Bc = dbl + (size_t)tok * 48 + 8;
        const float* Cc = dbl + (size_t)tok * 48 + 24;
        float dx = dtv * xv;
        float y = 0.f;
#pragma unroll
        for (int n = 0; n < NSTATE; ++n) {
            h[n] = expf(dtv * a[n]) * h[n] + dx * Bc[n];
            y += h[n] * Cc[n];
        }
        y += xv * Dd;
        float z = xz[(size_t)tok * 512 + 256 + d];
        ys[(size_t)tok * DINC + d] = y * siluf(z);
    }
}

// Scatter-add limb mamba output back into X1 at the limb's joints (residual).
__global__ void scatter_add_kernel(const float* __restrict__ proj, // (Ntok,128)
                                   float* __restrict__ X1,         // (972,17,128)
                                   int L, int jbase, int spine, int Ntok)
{
    int idx = blockIdx.x * blockDim.x + threadIdx.x;
    if (idx >= Ntok * DIMC) return;
    int tok = idx >> 7, c = idx & 127;
    int n = tok / L, l = tok - n * L;
    int joint = spine ? (l == 0 ? 0 : 6 + l) : (jbase + l);
    X1[(size_t)(n * NJ + joint) * DIMC + c] += proj[idx];
}

// Fusion MLP over joints (J=17 -> 34 -> 17), LN over J, in-place residual.
// One thread owns one (b*f, channel) column: no cross-thread hazard.
__global__ void fusion_kernel(float* __restrict__ X1,
                              const float* __restrict__ g,  const float* __restrict__ b,
                              const float* __restrict__ w1, const float* __restrict__ b1, // (34,17),(34)
                              const float* __restrict__ w2, const float* __restrict__ b2) // (17,34),(17)
{
    int idx = blockIdx.x * blockDim.x + threadIdx.x;
    if (idx >= NBF * DIMC) return;
    int bf = idx >> 7, c = idx & 127;
    float* col = X1 + (size_t)bf * NJ * DIMC + c;

    float v[NJ], vn[NJ];
    float m = 0.f;
#pragma unroll
    for (int j = 0; j < NJ; ++j) { v[j] = col[j * DIMC]; m += v[j]; }
    m *= (1.f / NJ);
    float var = 0.f;
#pragma unroll
    for (int j = 0; j < NJ; ++j) { float d = v[j] - m; var += d * d; }
    var *= (1.f / NJ);
    float inv = rsqrtf(var + 1e-5f);
#pragma unroll
    for (int j = 0; j < NJ; ++j) vn[j] = (v[j] - m) * inv * g[j] + b[j];

    float h[34];
    for (int k = 0; k < 34; ++k) {
        float acc = b1[k];
#pragma unroll
        for (int j = 0; j < NJ; ++j) acc += w1[k * NJ + j] * vn[j];
        h[k] = geluf(acc);
    }
#pragma unroll
    for (int j = 0; j < NJ; ++j) {
        float acc = b2[j];
        for (int k = 0; k < 34; ++k) acc += w2[j * 34 + k] * h[k];
        col[j * DIMC] = v[j] + acc;
    }
}

// (b,f,j,c) -> (b,j,f,c) token layout for the temporal mamba.
__global__ void transpose_kernel(const float* __restrict__ X1, float* __restrict__ XT)
{
    int idx = blockIdx.x * blockDim.x + threadIdx.x;
    if (idx >= NBF * NJ * DIMC) return;
    int c = idx & 127;
    int j = (idx >> 7) % NJ;
    int bf = idx / (NJ * DIMC);
    int bb = bf / NF, f = bf - bb * NF;
    XT[((size_t)(bb * NJ + j) * NF + f) * DIMC + c] = X1[idx];
}

__global__ void add_rows_kernel(float* __restrict__ dst, const float* __restrict__ src, int n)
{
    int idx = blockIdx.x * blockDim.x + threadIdx.x;
    if (idx < n) dst[idx] += src[idx];
}

// out[b,f,j,c] = XT[(b*17+j)*243+f, c] + mlp_out[same row, c]
__global__ void final_out_kernel(const float* __restrict__ XT,
                                 const float* __restrict__ Y,
                                 float* __restrict__ out)
{
    int idx = blockIdx.x * blockDim.x + threadIdx.x;
    if (idx >= NTOK_T * DIMC) return;
    int c = idx & 127;
    int f = (idx >> 7) % NF;
    int bj = idx / (NF * DIMC);
    int bb = bj / NJ, j = bj - bb * NJ;
    out[((size_t)(bb * NF + f) * NJ + j) * DIMC + c] = XT[idx] + Y[idx];
}

// ---------------------------------------------------------------------------
static void launch_gemm(int act, const float* A, int lda, const float* W, int ldw,
                        const float* bias, float* C, int ldc,
                        int M, int N, int K, hipStream_t stream)
{
    int mt  = (M + 15) >> 4;
    int nt2 = (N + 31) >> 5;               // 32-wide column pairs per wave
    long long waves = (long long)mt * nt2;
    int blocks = (int)((waves + 3) / 4);   // 4 waves (128 threads) per block
    dim3 grid(blocks), block(128);
    switch (act) {
        case 1:  wmma_gemm_kernel<1><<<grid, block, 0, stream>>>(A, lda, W, ldw, bias, C, ldc, M, N, K, nt2); break;
        case 2:  wmma_gemm_kernel<2><<<grid, block, 0, stream>>>(A, lda, W, ldw, bias, C, ldc, M, N, K, nt2); break;
        default: wmma_gemm_kernel<0><<<grid, block, 0, stream>>>(A, lda, W, ldw, bias, C, ldc, M, N, K, nt2); break;
    }
}

static inline int ceil_div(int a, int b) { return (a + b - 1) / b; }

extern "C" void kernel_launch(void* const* d_in, const int* in_sizes, int n_in,
                              void* d_out, int out_size, void* d_ws, size_t ws_size,
                              hipStream_t stream)
{
    (void)in_sizes; (void)n_in; (void)out_size; (void)ws_size;
    // Leaf order: x | 6 mamba tuples (9 each) | 8 ln pairs | fusion(4) | mlpT(4)
    const float* x = (const float*)d_in[0];
    auto P = [&](int i) { return (const float*)d_in[i]; };

    float* ws  = (float*)d_ws;
    float* X1  = ws;                         // (972,17,128)
    float* XT  = X1  + (size_t)NBF * NJ * DIMC; // (68,243,128)
    float* lnb = XT  + (size_t)NTOK_T * DIMC;   // NTPAD*128
    float* xzb = lnb + (size_t)NTPAD * DIMC;    // NTPAD*512 (also mlpT hidden)
    float* xcb = xzb + (size_t)NTPAD * 512;     // NTPAD*256
    float* dbb = xcb + (size_t)NTPAD * DINC;    // NTPAD*48
    float* dtb = dbb + (size_t)NTPAD * 48;      // NTPAD*256
    float* ysb = dtb + (size_t)NTPAD * DINC;    // NTPAD*256
    float* prb = ysb + (size_t)NTPAD * DINC;    // NTPAD*128

    // X1 = x  (concat[INV] restores original joint order -> residual in place)
    hipMemcpyAsync(X1, x, sizeof(float) * (size_t)NBF * NJ * DIMC,
                   hipMemcpyDeviceToDevice, stream);

    // --- Stage A: five limb mamba blocks (dconv=2) ---------------------------
    struct LimbDesc { int p, ln, L, jbase, spine; };
    const LimbDesc limbs[5] = {
        { 1, 55, 3,  1, 0},  // LL: joints 1..3,  ln1
        {10, 57, 3,  4, 0},  // RL: joints 4..6,  ln2
        {19, 59, 3, 14, 0},  // LA: joints 14..16, ln3
        {28, 61, 3, 11, 0},  // RA: joints 11..13, ln4
        {37, 63, 5,  0, 1},  // Sp: joints {0,7,8,9,10}, ln5
    };
    for (int i = 0; i < 5; ++i) {
        const LimbDesc& lb = limbs[i];
        int Ntok = NBF * lb.L;
        const float* in_w  = P(lb.p + 0);
        const float* convw = P(lb.p + 1);
        const float* convb = P(lb.p + 2);
        const float* xp_w  = P(lb.p + 3);
        const float* dt_w  = P(lb.p + 4);
        const float* dt_b  = P(lb.p + 5);
        const float* A_log = P(lb.p + 6);
        const float* Dpv   = P(lb.p + 7);
        const float* out_w = P(lb.p + 8);

        ln_kernel<<<Ntok, DIMC, 0, stream>>>(x, P(lb.ln), P(lb.ln + 1), lnb,
                                             lb.L, lb.jbase, lb.spine, 0, Ntok);
        launch_gemm(0, lnb, 128, in_w, 128, nullptr, xzb, 512, Ntok, 512, 128, stream);
        conv_silu_kernel<<<ceil_div(Ntok * DINC, 256), 256, 0, stream>>>(
            xzb, convw, convb, xcb, lb.L, 2, Ntok);
        launch_gemm(0, xcb, 256, xp_w, 256, nullptr, dbb, 48, Ntok, 40, 256, stream);
        launch_gemm(1, dbb, 48, dt_w, 8, dt_b, dtb, 256, Ntok, 256, 8, stream);
        scan_kernel<<<ceil_div(NBF * DINC, 256), 256, 0, stream>>>(
            dtb, dbb, xcb, xzb, A_log, Dpv, ysb, lb.L, NBF);
        launch_gemm(0, ysb, 256, out_w, 256, nullptr, prb, 128, Ntok, 128, 256, stream);
        scatter_add_kernel<<<ceil_div(Ntok * DIMC, 256), 256, 0, stream>>>(
            prb, X1, lb.L, lb.jbase, lb.spine, Ntok);
    }

    // --- Stage B: fusion MLP over joints (LN over J, residual, in place) ----
    fusion_kernel<<<ceil_div(NBF * DIMC, 256), 256, 0, stream>>>(
        X1, P(65), P(66), P(71), P(72), P(73), P(74));

    // --- Stage C: temporal mamba (dconv=4, L=243) + mlpT ---------------------
    transpose_kernel<<<ceil_div(NBF * NJ * DIMC, 256), 256, 0, stream>>>(X1, XT);
    {
        const int Ntok = NTOK_T, NB = NBATCH * NJ, p = 46;
        ln_kernel<<<Ntok, DIMC, 0, stream>>>(XT, P(67), P(68), lnb, 0, 0, 0, 1, Ntok);
        launch_gemm(0, lnb, 128, P(p + 0), 128, nullptr, xzb, 512, Ntok, 512, 128, stream);
        conv_silu_kernel<<<ceil_div(Ntok * DINC, 256), 256, 0, stream>>>(
            xzb, P(p + 1), P(p + 2), xcb, NF, 4, Ntok);
        launch_gemm(0, xcb, 256, P(p + 3), 256, nullptr, dbb, 48, Ntok, 40, 256, stream);
        launch_gemm(1, dbb, 48, P(p + 4), 8, P(p + 5), dtb, 256, Ntok, 256, 8, stream);
        scan_kernel<<<ceil_div(NB * DINC, 256), 256, 0, stream>>>(
            dtb, dbb, xcb, xzb, P(p + 6), P(p + 7), ysb, NF, NB);
        launch_gemm(0, ysb, 256, P(p + 8), 256, nullptr, prb, 128, Ntok, 128, 256, stream);
        add_rows_kernel<<<ceil_div(Ntok * DIMC, 256), 256, 0, stream>>>(XT, prb, Ntok * DIMC);

        // mlpT: XT + W2 @ gelu(W1 @ LN(XT))
        ln_kernel<<<Ntok, DIMC, 0, stream>>>(XT, P(69), P(70), lnb, 0, 0, 0, 1, Ntok);
        launch_gemm(2, lnb, 128, P(75), 128, P(76), xzb, 512, Ntok, 512, 128, stream);
        launch_gemm(0, xzb, 512, P(77), 512, P(78), prb, 128, Ntok, 128, 512, stream);
        final_out_kernel<<<ceil_div(Ntok * DIMC, 256), 256, 0, stream>>>(
            XT, prb, (float*)d_out);
    }
}